// GCN_pred_15513421873661
// MI455X (gfx1250) — compile-verified
//
#include <hip/hip_runtime.h>
#include <math.h>

// ---------------------------------------------------------------------------
// GCN inference for MI455X (gfx1250, wave32).
//   layer: h = x @ W (WMMA f32 16x16x4), a = D^-1/2 A_hat D^-1/2 h + b, relu
//   pool:  out[g] = sigmoid(mean_nodes(relu(a2)) @ lin_w + lin_b)
// GEMM uses V_WMMA_F32_16X16X4_F32 with W transposed in LDS (stride 132 floats
// -> conflict-free float2 DS reads). Edge scatter uses global_atomic_add_f32,
// which stays resident in the 192MB L2 (accumulator = 25.6MB).
// ---------------------------------------------------------------------------

typedef __attribute__((ext_vector_type(2))) float v2f;
typedef __attribute__((ext_vector_type(8))) float v8f;

#define DFEAT 132  // padded LDS stride (x132: all 32 lanes hit distinct bank pairs)
#define DD    128

// ---- degree / normalization ------------------------------------------------
__global__ void k_deg_init(float* __restrict__ deg, int n) {
  int i = blockIdx.x * blockDim.x + threadIdx.x;
  if (i < n) deg[i] = 1.0f;  // self loop contributes 1
}

__global__ void k_deg_count(const int* __restrict__ col, float* __restrict__ deg, int e) {
  int i = blockIdx.x * blockDim.x + threadIdx.x;
  if (i < e)
    __hip_atomic_fetch_add(&deg[col[i]], 1.0f, __ATOMIC_RELAXED, __HIP_MEMORY_SCOPE_AGENT);
}

__global__ void k_dinv(float* __restrict__ deg, int n) {
  int i = blockIdx.x * blockDim.x + threadIdx.x;
  if (i < n) deg[i] = rsqrtf(deg[i]);  // deg >= 1 always (self loop)
}

// ---- dense GEMM: H[n x 128] = X[n x 128] @ W[128 x 128] (WMMA f32) ---------
// Block = 256 threads = 8 waves; each wave computes a 16-row x 128-col strip.
// N = 50000 is a multiple of 16, so no partial tiles.
template <bool RELU_IN>
__global__ void k_gemm_wmma(const float* __restrict__ X, const float* __restrict__ W,
                            float* __restrict__ H, int n) {
  extern __shared__ float sW[];  // W transposed: sW[ncol*DFEAT + k], 128*132*4 B

  const int tid = threadIdx.x;
  for (int i = tid; i < DD * DD; i += blockDim.x) {
    int k = i >> 7, ncol = i & (DD - 1);
    sW[ncol * DFEAT + k] = W[i];
  }
  __syncthreads();

  const int wave   = tid >> 5;
  const int lane   = tid & 31;
  const int m_base = blockIdx.x * 128 + wave * 16;
  if (m_base >= n) return;  // no more barriers below

  const int lhalf = lane >> 4;  // 0: lanes 0-15, 1: lanes 16-31
  const int l16   = lane & 15;

  v8f acc[8];
  const v8f zero = {0.f, 0.f, 0.f, 0.f, 0.f, 0.f, 0.f, 0.f};
#pragma unroll
  for (int t = 0; t < 8; ++t) acc[t] = zero;

  // A fragment rows: lane's M = l16; K pair = k0 + 2*lhalf .. +1  (ISA 16x4 layout)
  const float2* xrow = reinterpret_cast<const float2*>(X + (size_t)(m_base + l16) * DD);

#pragma unroll 4
  for (int k0 = 0; k0 < DD; k0 += 4) {
    float2 af = xrow[(k0 >> 1) + lhalf];
    if (RELU_IN) { af.x = fmaxf(af.x, 0.f); af.y = fmaxf(af.y, 0.f); }
    v2f a; a.x = af.x; a.y = af.y;
#pragma unroll
    for (int t = 0; t < 8; ++t) {
      // B fragment: lane's N = t*16 + l16; K pair contiguous in transposed LDS
      const float2 bf =
          *reinterpret_cast<const float2*>(&sW[(t * 16 + l16) * DFEAT + k0 + 2 * lhalf]);
      v2f b; b.x = bf.x; b.y = bf.y;
      acc[t] = __builtin_amdgcn_wmma_f32_16x16x4_f32(
          /*neg_a=*/false, a, /*neg_b=*/false, b,
          /*c_mod=*/(short)0, acc[t], /*reuse_a=*/false, /*reuse_b=*/false);
    }
  }

  // C/D layout: VGPR i -> row (i + 8*lhalf), col = l16 within each 16-col tile
  float* hrow = H + (size_t)m_base * DD;
#pragma unroll
  for (int t = 0; t < 8; ++t) {
#pragma unroll
    for (int i = 0; i < 8; ++i) {
      hrow[(size_t)(i + lhalf * 8) * DD + t * 16 + l16] = acc[t][i];
    }
  }
}

// ---- aggregation init: a[i] = h[i]*dinv[i]^2 + bias  (self loop + bias) ----
__global__ void k_agg_init(const float* __restrict__ h, const float* __restrict__ dinv,
                           const float* __restrict__ bias, float* __restrict__ a, int n) {
  int idx = blockIdx.x * blockDim.x + threadIdx.x;
  if (idx >= n * 32) return;
  int node = idx >> 5;
  int d    = (idx & 31) << 2;
  float s  = dinv[node];
  float w  = s * s;
  const float4 hv = *reinterpret_cast<const float4*>(h + (size_t)node * DD + d);
  const float4 bv = *reinterpret_cast<const float4*>(bias + d);
  float4 o;
  o.x = hv.x * w + bv.x; o.y = hv.y * w + bv.y;
  o.z = hv.z * w + bv.z; o.w = hv.w * w + bv.w;
  *reinterpret_cast<float4*>(a + (size_t)node * DD + d) = o;
}

// ---- edge scatter: a[c] += h[r] * dinv[r]*dinv[c]  (one wave per edge) -----
__global__ void k_agg_edges(const int* __restrict__ row, const int* __restrict__ col,
                            const float* __restrict__ dinv, const float* __restrict__ h,
                            float* __restrict__ a, int e) {
  int wid  = (blockIdx.x * blockDim.x + threadIdx.x) >> 5;
  int lane = threadIdx.x & 31;
  if (wid >= e) return;
  int r = row[wid], c = col[wid];
  float wgt = dinv[r] * dinv[c];
  int d = lane << 2;
  const float4 hv = *reinterpret_cast<const float4*>(h + (size_t)r * DD + d);
  float* ap = a + (size_t)c * DD + d;
  __hip_atomic_fetch_add(ap + 0, hv.x * wgt, __ATOMIC_RELAXED, __HIP_MEMORY_SCOPE_AGENT);
  __hip_atomic_fetch_add(ap + 1, hv.y * wgt, __ATOMIC_RELAXED, __HIP_MEMORY_SCOPE_AGENT);
  __hip_atomic_fetch_add(ap + 2, hv.z * wgt, __ATOMIC_RELAXED, __HIP_MEMORY_SCOPE_AGENT);
  __hip_atomic_fetch_add(ap + 3, hv.w * wgt, __ATOMIC_RELAXED, __HIP_MEMORY_SCOPE_AGENT);
}

// ---- pooled linear: per-node dot(relu(a2), lin_w), segment-sum by graph ----
__global__ void k_zero_gc(float* __restrict__ sums, float* __restrict__ cnts, int g) {
  int i = blockIdx.x * blockDim.x + threadIdx.x;
  if (i < g) { sums[i] = 0.f; cnts[i] = 0.f; }
}

__global__ void k_pool(const float* __restrict__ a2, const int* __restrict__ batch,
                       const float* __restrict__ lw, float* __restrict__ sums,
                       float* __restrict__ cnts, int n) {
  int wid  = (blockIdx.x * blockDim.x + threadIdx.x) >> 5;
  int lane = threadIdx.x & 31;
  if (wid >= n) return;
  int d = lane << 2;
  const float4 hv = *reinterpret_cast<const float4*>(a2 + (size_t)wid * DD + d);
  const float4 wv = *reinterpret_cast<const float4*>(lw + d);
  float s = fmaxf(hv.x, 0.f) * wv.x + fmaxf(hv.y, 0.f) * wv.y +
            fmaxf(hv.z, 0.f) * wv.z + fmaxf(hv.w, 0.f) * wv.w;
#pragma unroll
  for (int off = 16; off > 0; off >>= 1) s += __shfl_xor(s, off, 32);
  if (lane == 0) {
    int g = batch[wid];
    __hip_atomic_fetch_add(&sums[g], s, __ATOMIC_RELAXED, __HIP_MEMORY_SCOPE_AGENT);
    __hip_atomic_fetch_add(&cnts[g], 1.0f, __ATOMIC_RELAXED, __HIP_MEMORY_SCOPE_AGENT);
  }
}

__global__ void k_final(const float* __restrict__ sums, const float* __restrict__ cnts,
                        const float* __restrict__ lb, float* __restrict__ out, int g) {
  int i = blockIdx.x * blockDim.x + threadIdx.x;
  if (i < g) {
    float z = sums[i] / fmaxf(cnts[i], 1.0f) + lb[0];
    out[i]  = 1.0f / (1.0f + expf(-z));
  }
}

// ---------------------------------------------------------------------------
extern "C" void kernel_launch(void* const* d_in, const int* in_sizes, int n_in,
                              void* d_out, int out_size, void* d_ws, size_t ws_size,
                              hipStream_t stream) {
  const float* x     = (const float*)d_in[0];
  const float* w1    = (const float*)d_in[1];
  const float* b1    = (const float*)d_in[2];
  const float* w2    = (const float*)d_in[3];
  const float* b2    = (const float*)d_in[4];
  const float* lin_w = (const float*)d_in[5];
  const float* lin_b = (const float*)d_in[6];
  const int*   row   = (const int*)d_in[7];         // edge_index[0]
  const int    E     = in_sizes[7] / 2;
  const int*   col   = row + E;                     // edge_index[1]
  const int*   batch = (const int*)d_in[8];
  const int    N     = in_sizes[0] / DD;
  const int    G     = out_size;
  float*       out   = (float*)d_out;

  // workspace carve-up (all offsets 256B aligned)
  char*  ws   = (char*)d_ws;
  size_t off  = 0;
  float* dinv = (float*)(ws + off); off += ((size_t)N * 4 + 255) & ~(size_t)255;
  float* bufh = (float*)(ws + off); off += (size_t)N * DD * 4;   // multiple of 256
  float* bufa = (float*)(ws + off); off += (size_t)N * DD * 4;
  float* sums = (float*)(ws + off); off += ((size_t)G * 4 + 255) & ~(size_t)255;
  float* cnts = (float*)(ws + off);

  const int    TB       = 256;
  const dim3   blk(TB);
  const size_t lds_gemm = (size_t)DD * DFEAT * sizeof(float);  // 67,584 B (< 320KB WGP LDS)
  const int    g_nodes  = (N + TB - 1) / TB;
  const int    g_edges  = (E + TB - 1) / TB;
  const int    g_gemm   = (N + 127) / 128;              // 8 waves x 16 rows per block
  const int    g_vec    = (N * 32 + TB - 1) / TB;       // node x 32 (4 floats/thread)
  const int    g_wedge  = (E * 32 + TB - 1) / TB;       // one wave per edge
  const int    g_wnode  = (N * 32 + TB - 1) / TB;       // one wave per node

  // normalization: dinv = rsqrt(1 + in-degree)
  k_deg_init<<<g_nodes, blk, 0, stream>>>(dinv, N);
  k_deg_count<<<g_edges, blk, 0, stream>>>(col, dinv, E);
  k_dinv<<<g_nodes, blk, 0, stream>>>(dinv, N);

  // layer 1
  k_gemm_wmma<false><<<g_gemm, blk, lds_gemm, stream>>>(x, w1, bufh, N);
  k_agg_init<<<g_vec, blk, 0, stream>>>(bufh, dinv, b1, bufa, N);
  k_agg_edges<<<g_wedge, blk, 0, stream>>>(row, col, dinv, bufh, bufa, E);

  // layer 2 (relu of layer-1 output fused into GEMM A-load)
  k_gemm_wmma<true><<<g_gemm, blk, lds_gemm, stream>>>(bufa, w2, bufh, N);
  k_agg_init<<<g_vec, blk, 0, stream>>>(bufh, dinv, b2, bufa, N);
  k_agg_edges<<<g_wedge, blk, 0, stream>>>(row, col, dinv, bufh, bufa, E);

  // mean pool + linear + sigmoid (relu fused into pool)
  k_zero_gc<<<(G + TB - 1) / TB, blk, 0, stream>>>(sums, cnts, G);
  k_pool<<<g_wnode, blk, 0, stream>>>(bufa, batch, lin_w, sums, cnts, N);
  k_final<<<(G + TB - 1) / TB, blk, 0, stream>>>(sums, cnts, lin_b, out, G);
}